// PPIModel_61692910240011
// MI455X (gfx1250) — compile-verified
//
#include <hip/hip_runtime.h>
#include <math.h>

// ---------------- problem constants (from reference) ----------------
#define BATCH       4
#define N_PER       50000
#define DEG         32
#define NUM_RELS    8
#define NUM_BASES   4
#define N_LAYERS    2
#define E_TOTAL     (BATCH * N_PER * DEG)   // 6,400,000
#define N_TOTAL     (BATCH * N_PER)         // 200,000
#define OUT_DIM     1024

// GEMM tiling: M=16 (rows 0-3: x1 graphs, 4-7: x2 graphs, 8-15: zero pad).
// Each block owns TWO 16-wide N tiles (32 outputs) -> 2 independent WMMA
// accumulator chains per wave. K = 50000 = KSPLIT(25) * 4 waves * 500.
// KPER_WAVE = 500 = 25 outer iterations * (5-wide unrolled inner * step 4).
#define NTILE2      (OUT_DIM / 32)          // 32 blocks in N
#define KSPLIT      25
#define KPER_BLOCK  (N_PER / KSPLIT)        // 2000
#define KPER_WAVE   (KPER_BLOCK / 4)        // 500
#define KUNROLL     5

typedef __attribute__((ext_vector_type(2))) float v2f;
typedef __attribute__((ext_vector_type(8))) float v8f;

__device__ __forceinline__ void atomAddF(float* p, float v) {
    // non-returning fp32 atomic add, device scope -> global_atomic_add_f32
    (void)__hip_atomic_fetch_add(p, v, __ATOMIC_RELAXED, __HIP_MEMORY_SCOPE_AGENT);
}

// wgt[l*8+r] = sum_b w_comp[l][r][b] * bases[l][b]   (16 scalars)
__global__ void wgt_kernel(const float* __restrict__ w_comp,
                           const float* __restrict__ bases,
                           float* __restrict__ wgt) {
    int i = threadIdx.x;
    if (i < N_LAYERS * NUM_RELS) {
        int l = i >> 3, r = i & 7;
        float s = 0.f;
#pragma unroll
        for (int b = 0; b < NUM_BASES; ++b)
            s += w_comp[(l * NUM_RELS + r) * NUM_BASES + b] * bases[l * NUM_BASES + b];
        wgt[i] = s;
    }
}

__global__ void copy2_kernel(const float* __restrict__ a, const float* __restrict__ b,
                             float* __restrict__ da, float* __restrict__ db, int n) {
    int i = blockIdx.x * blockDim.x + threadIdx.x;
    if (i < n) { da[i] = a[i]; db[i] = b[i]; }
}

__global__ void zero2_kernel(float* __restrict__ a, float* __restrict__ b, int n) {
    int i = blockIdx.x * blockDim.x + threadIdx.x;
    if (i < n) { a[i] = 0.f; b[i] = 0.f; }
}

__global__ void zero_kernel(float* __restrict__ a, int n) {
    int i = blockIdx.x * blockDim.x + threadIdx.x;
    if (i < n) a[i] = 0.f;
}

// One pass over edges serves BOTH feature sets (halves edge-stream traffic).
// Node arrays (~800 KB) are L2-resident -> random gathers/atomics are cheap.
__global__ void edge_kernel(const int* __restrict__ src, const int* __restrict__ dst,
                            const int* __restrict__ etype, const float* __restrict__ norm,
                            const float* __restrict__ x1, const float* __restrict__ x2,
                            const float* __restrict__ wgt,  // 8 scalars for this layer
                            float* __restrict__ agg1, float* __restrict__ agg2, int E) {
    int i = blockIdx.x * blockDim.x + threadIdx.x;
    int stride = gridDim.x * blockDim.x;
    for (int e = i; e < E; e += stride) {
        int s = src[e];
        int d = dst[e];
        float w = wgt[etype[e]] * norm[e];
        atomAddF(&agg1[d], x1[s] * w);
        atomAddF(&agg2[d], x2[s] * w);
    }
}

// h = agg + biases[l] ; relu (hidden layer only) ; x = h + initial
__global__ void node_kernel(const float* __restrict__ agg1, const float* __restrict__ agg2,
                            const float* __restrict__ init1, const float* __restrict__ init2,
                            const float* __restrict__ biases, int l, int do_relu,
                            float* __restrict__ x1, float* __restrict__ x2, int n) {
    int i = blockIdx.x * blockDim.x + threadIdx.x;
    if (i < n) {
        float bias = biases[l];
        float h1 = agg1[i] + bias;
        float h2 = agg2[i] + bias;
        if (do_relu) { h1 = fmaxf(h1, 0.f); h2 = fmaxf(h2, 0.f); }
        x1[i] = h1 + init1[i];
        x2[i] = h2 + init2[i];
    }
}

// D = A(16x4) x B(4x16) + C via V_WMMA_F32_16X16X4_F32, K streamed.
// A row m: m<4 -> x1 graph m ; m<8 -> x2 graph m-4 ; else zero (masked).
// B(k,n) = W_lin[n][k] -> per-lane float2 loads are K-contiguous for A and B.
// Two N tiles per block => two independent accumulator chains per wave; the
// inner constant-trip loop fully unrolls => 10 static WMMAs + 15 b64 loads
// per body so the scheduler can cluster loads and hide XDL latency.
// grid = NTILE2 * KSPLIT blocks, 128 threads (4 waves) per block.
__global__ void gemv_wmma_kernel(const float* __restrict__ x1, const float* __restrict__ x2,
                                 const float* __restrict__ W, float* __restrict__ g) {
    const int ntile = blockIdx.x / KSPLIT;      // 0..31 (32-wide output tile)
    const int kblk  = blockIdx.x % KSPLIT;
    const int nbase = ntile * 32;
    const int lane  = threadIdx.x & 31;
    const int wave  = threadIdx.x >> 5;

    const int m     = lane & 15;          // A row / B column (per CDNA5 ISA layouts)
    const int khalf = (lane >> 4) * 2;    // lanes 16-31 hold K+2,K+3

    const float* arow = (m < 4) ? (x1 + (size_t)m * N_PER)
                      : (m < 8) ? (x2 + (size_t)(m - 4) * N_PER)
                                : x1;                       // dummy, masked to 0
    const float amask = (m < 8) ? 1.0f : 0.0f;
    const float* brow0 = W + (size_t)(nbase + m) * N_PER;        // tile 0, K-contiguous
    const float* brow1 = W + (size_t)(nbase + 16 + m) * N_PER;   // tile 1

    const int k0 = kblk * KPER_BLOCK + wave * KPER_WAVE;

    v8f c0 = {0.f, 0.f, 0.f, 0.f, 0.f, 0.f, 0.f, 0.f};
    v8f c1 = {0.f, 0.f, 0.f, 0.f, 0.f, 0.f, 0.f, 0.f};
    for (int ko = 0; ko < KPER_WAVE / (4 * KUNROLL); ++ko) {
        const int kb = k0 + ko * 4 * KUNROLL;
        // speculative stream prefetch ~1000 floats ahead (global_prefetch_b8;
        // OOB translation failures silently dropped per ISA prefetch rules)
        __builtin_prefetch(brow0 + kb + 1000, 0, 0);
        __builtin_prefetch(brow1 + kb + 1000, 0, 0);
#pragma unroll
        for (int u = 0; u < KUNROLL; ++u) {
            const int k = kb + u * 4;
            v2f a = *(const v2f*)(arow + k + khalf);
            a *= amask;
            v2f b0 = *(const v2f*)(brow0 + k + khalf);
            v2f b1 = *(const v2f*)(brow1 + k + khalf);
            // (neg_a, A, neg_b, B, c_mod, C, reuse_a, reuse_b)
            c0 = __builtin_amdgcn_wmma_f32_16x16x4_f32(false, a, false, b0, (short)0,
                                                       c0, false, false);
            c1 = __builtin_amdgcn_wmma_f32_16x16x4_f32(false, a, false, b1, (short)0,
                                                       c1, false, false);
        }
    }

    // cross-wave reduction through LDS. C/D layout: VGPR v -> M = v (+8 for lanes
    // 16-31), N = lane & 15.
    __shared__ float red[4][512];
    const int mofs = (lane >> 4) << 3;
#pragma unroll
    for (int v = 0; v < 8; ++v) {
        red[wave][(v + mofs) * 16 + (lane & 15)]       = c0[v];
        red[wave][256 + (v + mofs) * 16 + (lane & 15)] = c1[v];
    }
    __syncthreads();

    for (int i = threadIdx.x; i < 512; i += 128) {
        float s = red[0][i] + red[1][i] + red[2][i] + red[3][i];
        int t = i >> 8;                 // tile 0/1
        int M = (i & 255) >> 4, N = i & 15;
        atomAddF(&g[M * OUT_DIM + nbase + t * 16 + N], s);  // g pre-zeroed per launch
    }
}

// out[m] = sigmoid( sum_n (g[m][n]+b_lin[n]) * (g[m+4][n]+b_lin[n]) ), m = 0..3
__global__ void final_kernel(const float* __restrict__ g, const float* __restrict__ b_lin,
                             float* __restrict__ out) {
    __shared__ float red[BATCH][256];
    float acc[BATCH] = {0.f, 0.f, 0.f, 0.f};
    for (int n = threadIdx.x; n < OUT_DIM; n += 256) {
        float bl = b_lin[n];
#pragma unroll
        for (int mrow = 0; mrow < BATCH; ++mrow) {
            float a = g[mrow * OUT_DIM + n] + bl;
            float b = g[(mrow + 4) * OUT_DIM + n] + bl;
            acc[mrow] += a * b;
        }
    }
#pragma unroll
    for (int mrow = 0; mrow < BATCH; ++mrow) red[mrow][threadIdx.x] = acc[mrow];
    __syncthreads();
    for (int s = 128; s > 0; s >>= 1) {
        if (threadIdx.x < (unsigned)s)
#pragma unroll
            for (int mrow = 0; mrow < BATCH; ++mrow)
                red[mrow][threadIdx.x] += red[mrow][threadIdx.x + s];
        __syncthreads();
    }
    if (threadIdx.x < BATCH) {
        float v = red[threadIdx.x][0];
        out[threadIdx.x] = 1.0f / (1.0f + __expf(-v));
    }
}

extern "C" void kernel_launch(void* const* d_in, const int* in_sizes, int n_in,
                              void* d_out, int out_size, void* d_ws, size_t ws_size,
                              hipStream_t stream) {
    const float* feat1  = (const float*)d_in[0];
    const float* feat2  = (const float*)d_in[1];
    const int*   src    = (const int*)d_in[2];
    const int*   dst    = (const int*)d_in[3];
    const int*   etype  = (const int*)d_in[4];
    const float* norm   = (const float*)d_in[5];
    /* d_in[6] = num_nodes (compile-time constant here) */
    const float* bases  = (const float*)d_in[7];
    const float* w_comp = (const float*)d_in[8];
    const float* biases = (const float*)d_in[9];
    const float* W_lin  = (const float*)d_in[10];
    const float* b_lin  = (const float*)d_in[11];
    float* out = (float*)d_out;

    // workspace layout (floats): wgt[16] | x1 | x2 | agg1 | agg2 | g[16*1024]
    float* ws   = (float*)d_ws;
    float* wgt  = ws;
    float* x1   = wgt  + 16;
    float* x2   = x1   + N_TOTAL;
    float* agg1 = x2   + N_TOTAL;
    float* agg2 = agg1 + N_TOTAL;
    float* g    = agg2 + N_TOTAL;

    const int nb_nodes = (N_TOTAL + 255) / 256;
    const int nb_edges = (E_TOTAL + 255) / 256;

    wgt_kernel<<<1, 32, 0, stream>>>(w_comp, bases, wgt);
    copy2_kernel<<<nb_nodes, 256, 0, stream>>>(feat1, feat2, x1, x2, N_TOTAL);

    for (int l = 0; l < N_LAYERS; ++l) {
        zero2_kernel<<<nb_nodes, 256, 0, stream>>>(agg1, agg2, N_TOTAL);
        edge_kernel<<<nb_edges, 256, 0, stream>>>(src, dst, etype, norm,
                                                  x1, x2, wgt + l * NUM_RELS,
                                                  agg1, agg2, E_TOTAL);
        node_kernel<<<nb_nodes, 256, 0, stream>>>(agg1, agg2, feat1, feat2,
                                                  biases, l,
                                                  (l < N_LAYERS - 1) ? 1 : 0,
                                                  x1, x2, N_TOTAL);
    }

    zero_kernel<<<(16 * OUT_DIM + 255) / 256, 256, 0, stream>>>(g, 16 * OUT_DIM);
    gemv_wmma_kernel<<<NTILE2 * KSPLIT, 128, 0, stream>>>(x1, x2, W_lin, g);
    final_kernel<<<1, 256, 0, stream>>>(g, b_lin, out);
}